// GraphAttentionEmbedding_85203561218640
// MI455X (gfx1250) — compile-verified
//
#include <hip/hip_runtime.h>
#include <hip/hip_bf16.h>
#include <math.h>

// ---------------------------------------------------------------------------
// CDNA5 / gfx1250 implementation of a 2-layer TransformerConv GNN.
//   Layer1: 128 -> 32 x 4 heads (concat -> 128), edge_dim 32
//   Layer2: 128 -> 64 x 1 head  (mean  -> 64),  edge_dim 32
// Dense projections: fused 4-way GEMM on V_WMMA_F32_16X16X4_F32 (full f32),
// A tile staged once in LDS (padded, bank-conflict-free) and reused for all
// four weight matrices (q/k/v/skip).
// Edge passes are wave-per-edge (wave32); the edge embedding is recomputed
// from LDS-resident We (vectorized ds reads) instead of materializing a
// 410 MB [E,128] tensor.
// ---------------------------------------------------------------------------

typedef float v2f __attribute__((ext_vector_type(2)));
typedef float v4f __attribute__((ext_vector_type(4)));
typedef float v8f __attribute__((ext_vector_type(8)));

// --------------------------- fused 4-way WMMA GEMM -------------------------
// O_w[M,NC] = A[M,K] @ W_w[K,NC] + b_w  for w = 0..3.
// One wave per 16x16 tile column; A tile staged in LDS with +4 float padding
// per row (row stride K+4 -> banks 4*r for r=0..15, conflict-free).
template <int K, int NC>
__global__ __launch_bounds__(32 * (NC / 16))
void gemm4_wmma_f32(const float* __restrict__ A,
                    const float* __restrict__ W0, const float* __restrict__ W1,
                    const float* __restrict__ W2, const float* __restrict__ W3,
                    const float* __restrict__ b0, const float* __restrict__ b1,
                    const float* __restrict__ b2, const float* __restrict__ b3,
                    float* __restrict__ O0, float* __restrict__ O1,
                    float* __restrict__ O2, float* __restrict__ O3, int M)
{
    constexpr int STRIDE = K + 4;           // padded row stride (floats)
    __shared__ float sA[16 * STRIDE];

    const int mtile = blockIdx.x;

    // cooperative load of the 16 x K A tile (float4 chunks, rows clamped)
    for (int t = threadIdx.x; t < 16 * (K / 4); t += blockDim.x) {
        const int r  = t / (K / 4);
        const int cc = t % (K / 4);
        int gr = mtile * 16 + r;
        if (gr >= M) gr = M - 1;
        *(float4*)&sA[r * STRIDE + cc * 4] =
            ((const float4*)(A + (size_t)gr * K))[cc];
    }
    __syncthreads();

    const int warp  = threadIdx.x >> 5;
    const int lane  = threadIdx.x & 31;
    const int lcol  = lane & 15;
    const int col   = warp * 16 + lcol;     // output column
    const int khalf = (lane >> 4) * 2;      // K split across lane halves
    const int rbase = mtile * 16 + 8 * (lane >> 4);
    const bool full = (mtile * 16 + 16) <= M;

    const float* __restrict__ Ws[4] = {W0, W1, W2, W3};
    const float* __restrict__ Bs[4] = {b0, b1, b2, b3};
    float* __restrict__       Os[4] = {O0, O1, O2, O3};

    #pragma unroll
    for (int w = 0; w < 4; ++w) {
        const float* __restrict__ B = Ws[w];
        v8f acc = {};
        #pragma unroll 4
        for (int k0 = 0; k0 < K; k0 += 4) {
            const int ka = k0 + khalf;
            v2f a, b;
            a[0] = sA[lcol * STRIDE + ka];
            a[1] = sA[lcol * STRIDE + ka + 1];
            b[0] = B[(size_t)ka * NC + col];
            b[1] = B[(size_t)(ka + 1) * NC + col];
            acc = __builtin_amdgcn_wmma_f32_16x16x4_f32(
                /*neg_a=*/false, a, /*neg_b=*/false, b,
                /*c_mod=*/(short)0, acc, /*reuse_a=*/false, /*reuse_b=*/false);
        }
        const float bv = Bs[w][col];
        float* __restrict__ O = Os[w];
        if (full) {
            #pragma unroll
            for (int i = 0; i < 8; ++i)
                O[(size_t)(rbase + i) * NC + col] = acc[i] + bv;
        } else {
            #pragma unroll
            for (int i = 0; i < 8; ++i) {
                const int r = rbase + i;
                if (r < M) O[(size_t)r * NC + col] = acc[i] + bv;
            }
        }
    }
}

// ------------------------------ small helpers ------------------------------
__global__ void fill_kernel(float* __restrict__ p, long long n, float v)
{
    long long i = (long long)blockIdx.x * blockDim.x + threadIdx.x;
    if (i < n) p[i] = v;
}

__global__ void relu_kernel(float* __restrict__ p, long long n)
{
    long long i = (long long)blockIdx.x * blockDim.x + threadIdx.x;
    if (i < n) p[i] = fmaxf(p[i], 0.0f);
}

__device__ __forceinline__ void atomicMaxF(float* addr, float val)
{
    unsigned int* ua  = (unsigned int*)addr;
    unsigned int  old = *ua;
    while (true) {
        if (__uint_as_float(old) >= val) break;
        unsigned int assumed = old;
        old = atomicCAS(ua, assumed, __float_as_uint(val));
        if (old == assumed) break;
    }
}

// vectorized LDS row read: CPL consecutive floats at 4*CPL-byte alignment
template <int CPL>
__device__ __forceinline__ void lds_row(const float* __restrict__ s, float* r)
{
    if constexpr (CPL == 4) {
        const v4f t = *(const v4f*)s;
        r[0] = t[0]; r[1] = t[1]; r[2] = t[2]; r[3] = t[3];
    } else if constexpr (CPL == 2) {
        const v2f t = *(const v2f*)s;
        r[0] = t[0]; r[1] = t[1];
    } else {
        #pragma unroll
        for (int i = 0; i < CPL; ++i) r[i] = s[i];
    }
}

// ----------------------------- edge kernels --------------------------------
// One wave per edge. C = H*D channels; each lane owns CPL = C/32 consecutive
// channels. EDGE_DIM==32: lane j holds ef[j]; broadcast via __shfl (wave32).
// We (32 x C) staged in LDS once per block.

template <int H, int D>
__global__ __launch_bounds__(256)
void edge_score(const float* __restrict__ q, const float* __restrict__ k,
                const float* __restrict__ ef, const float* __restrict__ We,
                const int* __restrict__ ei, int E,
                float* __restrict__ a, float* __restrict__ amax)
{
    constexpr int C   = H * D;
    constexpr int CPL = C / 32;      // channels per lane
    constexpr int G   = D / CPL;     // lanes per head (8 for L1, 32 for L2)
    __shared__ float sWe[32 * C];
    for (int i = threadIdx.x; i < 32 * C; i += blockDim.x) sWe[i] = We[i];
    __syncthreads();

    const int warp = threadIdx.x >> 5;
    const int lane = threadIdx.x & 31;
    const int e    = blockIdx.x * 8 + warp;
    if (e >= E) return;

    const int src = ei[e];
    const int dst = ei[E + e];
    const int c0  = lane * CPL;

    // edge embedding ee[c0..c0+CPL) = sum_j ef[e,j] * We[j, c]
    const float efl = ef[(size_t)e * 32 + lane];
    float ee[CPL];
    #pragma unroll
    for (int i = 0; i < CPL; ++i) ee[i] = 0.0f;
    #pragma unroll 8
    for (int j = 0; j < 32; ++j) {
        const float efj = __shfl(efl, j, 32);
        float wr[CPL];
        lds_row<CPL>(&sWe[j * C + c0], wr);
        #pragma unroll
        for (int i = 0; i < CPL; ++i) ee[i] += efj * wr[i];
    }

    // score partial: q[dst] . (k[src] + ee)
    const float* __restrict__ qd = q + (size_t)dst * C + c0;
    const float* __restrict__ ks = k + (size_t)src * C + c0;
    float p = 0.0f;
    #pragma unroll
    for (int i = 0; i < CPL; ++i) p += qd[i] * (ks[i] + ee[i]);

    // reduce within each head's G-lane aligned group
    #pragma unroll
    for (int m = 1; m < G; m <<= 1) p += __shfl_xor(p, m, 32);

    if ((lane & (G - 1)) == 0) {
        const int   h = c0 / D;
        const float s = p * rsqrtf((float)D);
        a[(size_t)e * H + h] = s;
        atomicMaxF(&amax[(size_t)dst * H + h], s);
    }
}

template <int H>
__global__ __launch_bounds__(256)
void edge_exp(float* __restrict__ a, const float* __restrict__ amax,
              const int* __restrict__ ei, int E, float* __restrict__ den)
{
    long long t = (long long)blockIdx.x * blockDim.x + threadIdx.x;
    if (t >= (long long)E * H) return;
    const int e  = (int)(t / H);
    const int h  = (int)(t % H);
    const int d  = ei[E + e];
    const float ex = __expf(a[t] - amax[(size_t)d * H + h]);
    a[t] = ex;
    atomicAdd(&den[(size_t)d * H + h], ex);
}

template <int H, int D>
__global__ __launch_bounds__(256)
void edge_aggr(const float* __restrict__ v, const float* __restrict__ ef,
               const float* __restrict__ We, const int* __restrict__ ei, int E,
               const float* __restrict__ aex, const float* __restrict__ den,
               float* __restrict__ out)
{
    constexpr int C   = H * D;
    constexpr int CPL = C / 32;
    __shared__ float sWe[32 * C];
    for (int i = threadIdx.x; i < 32 * C; i += blockDim.x) sWe[i] = We[i];
    __syncthreads();

    const int warp = threadIdx.x >> 5;
    const int lane = threadIdx.x & 31;
    const int e    = blockIdx.x * 8 + warp;
    if (e >= E) return;

    const int src = ei[e];
    const int dst = ei[E + e];
    const int c0  = lane * CPL;

    const float efl = ef[(size_t)e * 32 + lane];
    float ee[CPL];
    #pragma unroll
    for (int i = 0; i < CPL; ++i) ee[i] = 0.0f;
    #pragma unroll 8
    for (int j = 0; j < 32; ++j) {
        const float efj = __shfl(efl, j, 32);
        float wr[CPL];
        lds_row<CPL>(&sWe[j * C + c0], wr);
        #pragma unroll
        for (int i = 0; i < CPL; ++i) ee[i] += efj * wr[i];
    }

    const int   h     = c0 / D;
    const float alpha = aex[(size_t)e * H + h] / den[(size_t)dst * H + h];
    const float* __restrict__ vs = v + (size_t)src * C + c0;
    #pragma unroll
    for (int i = 0; i < CPL; ++i)
        atomicAdd(&out[(size_t)dst * C + c0 + i], alpha * (vs[i] + ee[i]));
}

// ------------------------------- launcher ----------------------------------
extern "C" void kernel_launch(void* const* d_in, const int* in_sizes, int n_in,
                              void* d_out, int out_size, void* d_ws, size_t ws_size,
                              hipStream_t stream)
{
    const float* x    = (const float*)d_in[0];
    const int*   ei   = (const int*)d_in[1];
    const float* ef   = (const float*)d_in[2];
    const float* Wq1  = (const float*)d_in[3];
    const float* bq1  = (const float*)d_in[4];
    const float* Wk1  = (const float*)d_in[5];
    const float* bk1  = (const float*)d_in[6];
    const float* Wv1  = (const float*)d_in[7];
    const float* bv1  = (const float*)d_in[8];
    const float* We1  = (const float*)d_in[9];
    const float* Ws1  = (const float*)d_in[10];
    const float* bs1  = (const float*)d_in[11];
    const float* Wq2  = (const float*)d_in[12];
    const float* bq2  = (const float*)d_in[13];
    const float* Wk2  = (const float*)d_in[14];
    const float* bk2  = (const float*)d_in[15];
    const float* Wv2  = (const float*)d_in[16];
    const float* bv2  = (const float*)d_in[17];
    const float* We2  = (const float*)d_in[18];
    const float* Ws2  = (const float*)d_in[19];
    const float* bs2  = (const float*)d_in[20];

    const int N = in_sizes[0] / 128;
    const int E = in_sizes[1] / 2;

    // workspace layout (floats)
    float* ws    = (float*)d_ws;
    float* q1    = ws;                      // N*128
    float* k1    = q1 + (size_t)N * 128;    // N*128
    float* v1    = k1 + (size_t)N * 128;    // N*128
    float* h1    = v1 + (size_t)N * 128;    // N*128  (skip proj, then += attn)
    float* a1    = h1 + (size_t)N * 128;    // E*4
    float* amax1 = a1 + (size_t)E * 4;      // N*4
    float* den1  = amax1 + (size_t)N * 4;   // N*4
    float* q2    = den1 + (size_t)N * 4;    // N*64
    float* k2    = q2 + (size_t)N * 64;     // N*64
    float* v2    = k2 + (size_t)N * 64;     // N*64
    float* a2    = v2 + (size_t)N * 64;     // E
    float* amax2 = a2 + (size_t)E;          // N
    float* den2  = amax2 + (size_t)N;       // N
    float* out   = (float*)d_out;           // N*64

    const int mtiles = (N + 15) / 16;
    const int eblk   = (E + 7) / 8;
    const float NEG_INF = -__builtin_inff();

    // ---- layer 1: fused q/k/v/skip projections (f32 WMMA, shared A tile) ----
    gemm4_wmma_f32<128, 128><<<mtiles, 256, 0, stream>>>(
        x, Wq1, Wk1, Wv1, Ws1, bq1, bk1, bv1, bs1, q1, k1, v1, h1, N);

    // ---- layer 1 segment softmax + aggregation ----
    fill_kernel<<<((long long)N * 4 + 255) / 256, 256, 0, stream>>>(amax1, (long long)N * 4, NEG_INF);
    fill_kernel<<<((long long)N * 4 + 255) / 256, 256, 0, stream>>>(den1,  (long long)N * 4, 0.0f);
    edge_score<4, 32><<<eblk, 256, 0, stream>>>(q1, k1, ef, We1, ei, E, a1, amax1);
    edge_exp<4><<<((long long)E * 4 + 255) / 256, 256, 0, stream>>>(a1, amax1, ei, E, den1);
    edge_aggr<4, 32><<<eblk, 256, 0, stream>>>(v1, ef, We1, ei, E, a1, den1, h1);
    relu_kernel<<<((long long)N * 128 + 255) / 256, 256, 0, stream>>>(h1, (long long)N * 128);

    // ---- layer 2: fused q/k/v/skip projections ----
    gemm4_wmma_f32<128, 64><<<mtiles, 128, 0, stream>>>(
        h1, Wq2, Wk2, Wv2, Ws2, bq2, bk2, bv2, bs2, q2, k2, v2, out, N);

    // ---- layer 2 segment softmax + aggregation (heads=1, D=64) ----
    fill_kernel<<<((long long)N + 255) / 256, 256, 0, stream>>>(amax2, (long long)N, NEG_INF);
    fill_kernel<<<((long long)N + 255) / 256, 256, 0, stream>>>(den2,  (long long)N, 0.0f);
    edge_score<1, 64><<<eblk, 256, 0, stream>>>(q2, k2, ef, We2, ei, E, a2, amax2);
    edge_exp<1><<<((long long)E + 255) / 256, 256, 0, stream>>>(a2, amax2, ei, E, den2);
    edge_aggr<1, 64><<<eblk, 256, 0, stream>>>(v2, ef, We2, ei, E, a2, den2, out);
}